// CGMM_41111426957570
// MI455X (gfx1250) — compile-verified
//
#include <hip/hip_runtime.h>
#include <hip/hip_bf16.h>

// CGMM two-layer forward for MI455X (gfx1250, wave32).
// d_in: x[N] i32, edge_index[2,E] i32, B0[C,M,G] f32, Pi[C,G] f32,
//       B1[C,M,G] f32, Q_neigh[C,C,G] f32   (C=16, M=32, G=8)
// d_out: ll[N,2,G] f32  then  post1[N,C,G] f32  (flat concat)

typedef __attribute__((ext_vector_type(2))) float v2f;
typedef __attribute__((ext_vector_type(8))) float v8f;

#define CC 16
#define MM 32
#define GG 8

// ds_swizzle_b32 group-of-32 XOR patterns (offset[15]=0, and_mask=0x1F,
// xor_mask in bits[14:10]) -- single DS op, no bounds-check VALU like bpermute.
#define SWZ_XOR(v, imm) \
  __int_as_float(__builtin_amdgcn_ds_swizzle(__float_as_int(v), (imm)))

// ---------------------------------------------------------------------------
// Kernel 1: all four softmax tables (392 independent rows, each <= 32 elems)
// ---------------------------------------------------------------------------
__global__ __launch_bounds__(256) void cgmm_softmax(
    const float* __restrict__ B0, const float* __restrict__ Pi,
    const float* __restrict__ B1, const float* __restrict__ Qn,
    float* __restrict__ smB0, float* __restrict__ smPi,
    float* __restrict__ smB1, float* __restrict__ smQ) {
  int r = blockIdx.x * blockDim.x + threadIdx.x;
  if (r < 256) {
    // B0 / B1: softmax over M (axis=1); row = (c,g)
    const float* src = (r < 128) ? B0 : B1;
    float* dst = (r < 128) ? smB0 : smB1;
    int rr = r & 127;
    int c = rr >> 3, g = rr & 7;
    float mx = -1e30f;
    for (int m = 0; m < MM; ++m) mx = fmaxf(mx, src[(c * MM + m) * GG + g]);
    float se = 0.f;
    for (int m = 0; m < MM; ++m) se += __expf(src[(c * MM + m) * GG + g] - mx);
    float inv = 1.f / se;
    for (int m = 0; m < MM; ++m)
      dst[(c * MM + m) * GG + g] = __expf(src[(c * MM + m) * GG + g] - mx) * inv;
  } else if (r < 264) {
    // Pi: softmax over C (axis=0); row = g
    int g = r - 256;
    float mx = -1e30f;
    for (int c = 0; c < CC; ++c) mx = fmaxf(mx, Pi[c * GG + g]);
    float se = 0.f;
    for (int c = 0; c < CC; ++c) se += __expf(Pi[c * GG + g] - mx);
    float inv = 1.f / se;
    for (int c = 0; c < CC; ++c)
      smPi[c * GG + g] = __expf(Pi[c * GG + g] - mx) * inv;
  } else if (r < 392) {
    // Q_neigh: softmax over dim 0 (i); row = (l,g)
    int rr = r - 264;
    int l = rr >> 3, g = rr & 7;
    float mx = -1e30f;
    for (int i = 0; i < CC; ++i) mx = fmaxf(mx, Qn[(i * CC + l) * GG + g]);
    float se = 0.f;
    for (int i = 0; i < CC; ++i) se += __expf(Qn[(i * CC + l) * GG + g] - mx);
    float inv = 1.f / se;
    for (int i = 0; i < CC; ++i)
      smQ[(i * CC + l) * GG + g] = __expf(Qn[(i * CC + l) * GG + g] - mx) * inv;
  }
}

// ---------------------------------------------------------------------------
// Kernel 2: layer 0. One block = 2 nodes; thread (j,p): c=p>>3, g=p&7.
// Writes post0[N,C,G] to ws and ll0 -> out[n*16 + g].
// ---------------------------------------------------------------------------
__global__ __launch_bounds__(256) void cgmm_layer0(
    const int* __restrict__ x, const float* __restrict__ smB0,
    const float* __restrict__ smPi, float* __restrict__ post0,
    float* __restrict__ out, int N) {
  __shared__ float red[2][128];
  __shared__ float norms[2][8];
  int t = threadIdx.x;
  int j = t >> 7, p = t & 127;
  int c = p >> 3, g = p & 7;
  int node = blockIdx.x * 2 + j;
  if (node >= N) return;
  int xn = x[node];
  float u = smPi[c * GG + g] * smB0[(c * MM + xn) * GG + g];
  red[j][p] = u;
  __syncthreads();
  if (c == 0) {
    float s = 0.f;
    for (int cc = 0; cc < CC; ++cc) s += red[j][cc * GG + g];
    norms[j][g] = s;
    out[(size_t)node * 16 + g] = logf(s);  // ll0 at [n,0,g]
  }
  __syncthreads();
  post0[(size_t)node * 128 + p] = u / norms[j][g];
}

// ---------------------------------------------------------------------------
// Kernel 3: edge scatter-add. One wave32 per edge; lane moves float4 (16 B)
// -> coalesced 512 B gather of post0[dst], L2-resident f32 atomic adds into
// aggr[src] (25.6 MB accumulator lives in the 192 MB L2). cnt[src] += 1.
// ---------------------------------------------------------------------------
__global__ __launch_bounds__(256) void cgmm_edges(
    const int* __restrict__ edge_index, const float* __restrict__ post0,
    float* __restrict__ aggr, float* __restrict__ cnt, int E) {
  int wave = blockIdx.x * 8 + (threadIdx.x >> 5);
  int lane = threadIdx.x & 31;
  if (wave >= E) return;
  int s = edge_index[wave];       // src (segment id)
  int d = edge_index[E + wave];   // dst (gathered)
  const float4* srcp = (const float4*)(post0 + (size_t)d * 128) + lane;
  float4 v = *srcp;
  float* dstp = aggr + (size_t)s * 128 + lane * 4;
  atomicAdd(dstp + 0, v.x);
  atomicAdd(dstp + 1, v.y);
  atomicAdd(dstp + 2, v.z);
  atomicAdd(dstp + 3, v.w);
  if (lane == 0) atomicAdd(cnt + s, 1.0f);
}

// ---------------------------------------------------------------------------
// Kernel 4: layer 1 with V_WMMA_F32_16X16X4_F32.
// Block = 256 threads = 8 waves; wave w handles g=w for a 16-node tile.
// S(16x16) = Aggr(16x16) @ Q^T(16x16) via 4 chained K=4 f32 WMMAs.
// Then post1 = B*S / sum_i(B*S), ll1 = log(sum).   EXEC all-ones at WMMAs.
// i-reduction uses ds_swizzle SWAPX1/2/4/8 (pure DS, no bpermute overhead).
// ---------------------------------------------------------------------------
__global__ __launch_bounds__(256) void cgmm_layer1(
    const int* __restrict__ x, const float* __restrict__ smB1,
    const float* __restrict__ smQ, const float* __restrict__ aggr_sum,
    const float* __restrict__ cnt, float* __restrict__ out, int N) {
  const int tid = threadIdx.x;
  const int g = tid >> 5;        // wave id == generative model id
  const int lane = tid & 31;
  const int half = lane >> 4;    // 0: lanes 0-15, 1: lanes 16-31
  const int col = lane & 15;     // column index (state i) / A-row (node m)
  const int base = blockIdx.x * 16;

  __shared__ int xs[16];
  if (tid < 16) xs[tid] = x[base + tid];
  __syncthreads();

  // ---- A operand: aggr (scatter-mean) rows; lane holds node m = col ----
  const float invc = 1.0f / fmaxf(cnt[base + col], 1.0f);
  const float* Arow = aggr_sum + (size_t)(base + col) * 128 + g;  // stride 8 per l

  v8f acc = {};
#pragma unroll
  for (int c4 = 0; c4 < 4; ++c4) {
    const int l0 = c4 * 4 + 2 * half;  // ISA A 16x4: lanes>=16 hold K+2
    v2f a, b;
    a.x = Arow[(l0 + 0) * 8] * invc;                 // A[m, l0]
    a.y = Arow[(l0 + 1) * 8] * invc;                 // A[m, l0+1]
    b.x = smQ[(col * CC + (l0 + 0)) * GG + g];       // B[l0,  i] = Q[i,l0,g]
    b.y = smQ[(col * CC + (l0 + 1)) * GG + g];       // B[l0+1,i]
    acc = __builtin_amdgcn_wmma_f32_16x16x4_f32(
        /*neg_a=*/false, a, /*neg_b=*/false, b,
        /*c_mod=*/(short)0, acc, /*reuse_a=*/false, /*reuse_b=*/false);
  }

  // ---- D layout: acc[v], lane -> node = base + v + 8*half, state i = col ----
  const size_t post1_base = (size_t)N * 16;  // ll block is N*2*G floats
#pragma unroll
  for (int v = 0; v < 8; ++v) {
    const int node = base + v + 8 * half;
    const int xn = xs[v + 8 * half];
    const float bn = smB1[(col * MM + xn) * GG + g];  // sm_B1[i, x[n], g]
    const float w = bn * acc[v];
    // sum over i (16 lanes in this half) -> norm1[node, g]
    float s = w;
    s += SWZ_XOR(s, 0x041F);  // SWAPX1
    s += SWZ_XOR(s, 0x081F);  // SWAPX2
    s += SWZ_XOR(s, 0x101F);  // SWAPX4
    s += SWZ_XOR(s, 0x201F);  // SWAPX8
    out[post1_base + (size_t)node * 128 + col * GG + g] = w / s;  // post1
    if (col == 0) out[(size_t)node * 16 + 8 + g] = logf(s);       // ll1 [n,1,g]
  }
}

// ---------------------------------------------------------------------------
extern "C" void kernel_launch(void* const* d_in, const int* in_sizes, int n_in,
                              void* d_out, int out_size, void* d_ws, size_t ws_size,
                              hipStream_t stream) {
  const int* x = (const int*)d_in[0];
  const int* edge_index = (const int*)d_in[1];
  const float* B0 = (const float*)d_in[2];
  const float* Pi = (const float*)d_in[3];
  const float* B1 = (const float*)d_in[4];
  const float* Qn = (const float*)d_in[5];
  float* out = (float*)d_out;

  const int N = in_sizes[0];
  const int E = in_sizes[1] / 2;

  // workspace layout (floats)
  float* ws = (float*)d_ws;
  float* smB0 = ws;                          // 4096
  float* smB1 = smB0 + CC * MM * GG;         // 4096
  float* smPi = smB1 + CC * MM * GG;         // 128
  float* smQ = smPi + CC * GG;               // 2048
  float* post0 = smQ + CC * CC * GG;         // N*128
  float* aggr = post0 + (size_t)N * 128;     // N*128
  float* cnt = aggr + (size_t)N * 128;       // N

  // zero the scatter accumulators (aggr + cnt are contiguous)
  hipMemsetAsync(aggr, 0, ((size_t)N * 128 + (size_t)N) * sizeof(float), stream);

  cgmm_softmax<<<2, 256, 0, stream>>>(B0, Pi, B1, Qn, smB0, smPi, smB1, smQ);
  cgmm_layer0<<<(N + 1) / 2, 256, 0, stream>>>(x, smB0, smPi, post0, out, N);
  cgmm_edges<<<(E + 7) / 8, 256, 0, stream>>>(edge_index, post0, aggr, cnt, E);
  cgmm_layer1<<<N / 16, 256, 0, stream>>>(x, smB1, smQ, aggr, cnt, out, N);
}